// DiscriminativeLoss_79757542686901
// MI455X (gfx1250) — compile-verified
//
#include <hip/hip_runtime.h>
#include <hip/hip_bf16.h>

typedef __attribute__((ext_vector_type(2))) float v2f;
typedef __attribute__((ext_vector_type(8))) float v8f;

#define NUM_LANES 5
#define DELTA_V 0.5f
#define DELTA_D 3.0f
#define HW (512 * 1024)
#define CCH 4

// Raw hardware v_sqrt_f32 (~1 ULP), avoids the IEEE refinement sequence.
__device__ __forceinline__ float fast_sqrtf(float x) { return __builtin_amdgcn_sqrtf(x); }

// ---------------------------------------------------------------------------
// Kernel 0: zero workspace accumulators
// ---------------------------------------------------------------------------
__global__ void dl_zero_ws(float* w, int n) {
    int i = blockIdx.x * blockDim.x + threadIdx.x;
    if (i < n) w[i] = 0.0f;
}

// ---------------------------------------------------------------------------
// Kernel 1: per-(batch,lane) counts and per-channel sums.
// grid = (128, B), block = 256. Streams the 64MB embedding + labels once.
// ---------------------------------------------------------------------------
__device__ __forceinline__ void accum_pix(int lbl, float e0, float e1, float e2, float e3,
                                          float cnt[NUM_LANES], float sum[NUM_LANES][CCH]) {
#pragma unroll
    for (int k = 0; k < NUM_LANES; ++k) {
        float m = (lbl == k + 1) ? 1.0f : 0.0f;
        cnt[k] += m;
        sum[k][0] = fmaf(m, e0, sum[k][0]);
        sum[k][1] = fmaf(m, e1, sum[k][1]);
        sum[k][2] = fmaf(m, e2, sum[k][2]);
        sum[k][3] = fmaf(m, e3, sum[k][3]);
    }
}

__global__ void dl_sums_kernel(const float* __restrict__ emb, const int* __restrict__ lab,
                               float* __restrict__ counts, float* __restrict__ sums) {
    const int b = blockIdx.y;
    const float* eb = emb + (size_t)b * CCH * HW;
    const int* lb = lab + (size_t)b * HW;

    float cnt[NUM_LANES] = {0.f, 0.f, 0.f, 0.f, 0.f};
    float sum[NUM_LANES][CCH] = {};

    const int tid = blockIdx.x * blockDim.x + threadIdx.x;
    const int nth = gridDim.x * blockDim.x;
    for (int p4 = tid; p4 < HW / 4; p4 += nth) {
        const int p = p4 * 4;
        int4 l = *(const int4*)(lb + p);
        float4 e0 = *(const float4*)(eb + 0 * HW + p);
        float4 e1 = *(const float4*)(eb + 1 * HW + p);
        float4 e2 = *(const float4*)(eb + 2 * HW + p);
        float4 e3 = *(const float4*)(eb + 3 * HW + p);
        accum_pix(l.x, e0.x, e1.x, e2.x, e3.x, cnt, sum);
        accum_pix(l.y, e0.y, e1.y, e2.y, e3.y, cnt, sum);
        accum_pix(l.z, e0.z, e1.z, e2.z, e3.z, cnt, sum);
        accum_pix(l.w, e0.w, e1.w, e2.w, e3.w, cnt, sum);
    }

    // wave32 shuffle reduction of 25 partials
#pragma unroll
    for (int k = 0; k < NUM_LANES; ++k) {
        for (int off = 16; off > 0; off >>= 1) cnt[k] += __shfl_down(cnt[k], off, 32);
#pragma unroll
        for (int c = 0; c < CCH; ++c)
            for (int off = 16; off > 0; off >>= 1) sum[k][c] += __shfl_down(sum[k][c], off, 32);
    }

    __shared__ float acc[25];
    if (threadIdx.x < 25) acc[threadIdx.x] = 0.0f;
    __syncthreads();
    if ((threadIdx.x & 31) == 0) {
#pragma unroll
        for (int k = 0; k < NUM_LANES; ++k) {
            atomicAdd(&acc[k], cnt[k]);
#pragma unroll
            for (int c = 0; c < CCH; ++c) atomicAdd(&acc[NUM_LANES + k * CCH + c], sum[k][c]);
        }
    }
    __syncthreads();
    if (threadIdx.x < NUM_LANES) {
        atomicAdd(&counts[b * NUM_LANES + threadIdx.x], acc[threadIdx.x]);
    } else if (threadIdx.x < 25) {
        int j = threadIdx.x - NUM_LANES;  // k*4 + c
        atomicAdd(&sums[b * NUM_LANES * CCH + j], acc[NUM_LANES + j]);
    }
}

// ---------------------------------------------------------------------------
// Kernel 2: centroids + pairwise term via V_WMMA_F32_16X16X4_F32 (Gram matrix,
// inner dim K=4 == channel count). Single wave32 block; one WMMA per batch.
// ---------------------------------------------------------------------------
__global__ void dl_means_gram_kernel(const float* __restrict__ sums,
                                     const float* __restrict__ counts,
                                     float* __restrict__ means, float* __restrict__ pvals) {
    __shared__ float sm_means[8][16][CCH];  // rows padded to 16, zero-filled
    __shared__ float sm_norm[8][16];
    const int lane = threadIdx.x;  // 0..31

    for (int i = lane; i < 8 * 16 * CCH; i += 32) ((float*)sm_means)[i] = 0.0f;
    __syncthreads();

    // means[b][k][c] = sums / counts  (160 values, linear layout matches)
    for (int i = lane; i < 8 * NUM_LANES * CCH; i += 32) {
        int b = i / (NUM_LANES * CCH);
        int r = i % (NUM_LANES * CCH);
        float m = sums[i] / counts[b * NUM_LANES + (r >> 2)];
        means[i] = m;
        sm_means[b][r >> 2][r & 3] = m;
    }
    __syncthreads();

    // squared norms per (padded) row; zero rows give 0
    for (int i = lane; i < 8 * 16; i += 32) {
        int b = i >> 4, r = i & 15;
        const float* mr = sm_means[b][r];
        sm_norm[b][r] = mr[0] * mr[0] + mr[1] * mr[1] + mr[2] * mr[2] + mr[3] * mr[3];
    }
    __syncthreads();

    const int row = lane & 15;
    const int half = lane >> 4;  // 0: cols/K 0..1 ; 1: cols/K 2..3 (A f32 16x4 layout)

    for (int b = 0; b < 8; ++b) {
        // A (16x4, M=row, K=2*half+v) and B (4x16, N=row, K=2*half+v) share registers
        v2f a;
        a.x = sm_means[b][row][2 * half + 0];
        a.y = sm_means[b][row][2 * half + 1];
        v8f cz = {};
        v8f g = __builtin_amdgcn_wmma_f32_16x16x4_f32(false, a, false, a, (short)0, cz,
                                                      false, false);
        // D element i of this lane: M = half*8 + i, N = row
        float nn = sm_norm[b][row];
        float psum = 0.0f;
        if (row < NUM_LANES) {
#pragma unroll
            for (int i = 0; i < 8; ++i) {
                int m = half * 8 + i;
                if (m < NUM_LANES) {
                    if (m != row) {
                        float d2 = fmaxf(sm_norm[b][m] + nn - 2.0f * g[i], 0.0f);
                        float t = fmaxf(DELTA_D - fast_sqrtf(d2), 0.0f);
                        psum += t * t;
                    }
                }
            }
        }
        for (int off = 16; off > 0; off >>= 1) psum += __shfl_down(psum, off, 32);
        if (lane == 0) pvals[b] = psum;
    }
}

// ---------------------------------------------------------------------------
// Kernel 3: variance term. Second streaming pass (L2-resident: 80MB < 192MB).
// ---------------------------------------------------------------------------
__device__ __forceinline__ void accum_var(int lbl, float e0, float e1, float e2, float e3,
                                          const float (*sm)[CCH], float acc[NUM_LANES]) {
    int k = lbl - 1;
    bool valid = (k >= 0) && (k < NUM_LANES);
    int kk = valid ? k : 0;
    float d0 = e0 - sm[kk][0];
    float d1 = e1 - sm[kk][1];
    float d2 = e2 - sm[kk][2];
    float d3 = e3 - sm[kk][3];
    float dd = d0 * d0 + d1 * d1 + d2 * d2 + d3 * d3;
    float t = fmaxf(fast_sqrtf(dd) - DELTA_V, 0.0f);
    float t2 = valid ? t * t : 0.0f;
#pragma unroll
    for (int k2 = 0; k2 < NUM_LANES; ++k2) acc[k2] += (lbl == k2 + 1) ? t2 : 0.0f;
}

__global__ void dl_var_kernel(const float* __restrict__ emb, const int* __restrict__ lab,
                              const float* __restrict__ means, float* __restrict__ varsum) {
    const int b = blockIdx.y;
    const float* eb = emb + (size_t)b * CCH * HW;
    const int* lb = lab + (size_t)b * HW;

    __shared__ float sm[NUM_LANES][CCH];
    if (threadIdx.x < NUM_LANES * CCH)
        ((float*)sm)[threadIdx.x] = means[b * NUM_LANES * CCH + threadIdx.x];
    __syncthreads();

    float acc[NUM_LANES] = {0.f, 0.f, 0.f, 0.f, 0.f};
    const int tid = blockIdx.x * blockDim.x + threadIdx.x;
    const int nth = gridDim.x * blockDim.x;
    for (int p4 = tid; p4 < HW / 4; p4 += nth) {
        const int p = p4 * 4;
        int4 l = *(const int4*)(lb + p);
        float4 e0 = *(const float4*)(eb + 0 * HW + p);
        float4 e1 = *(const float4*)(eb + 1 * HW + p);
        float4 e2 = *(const float4*)(eb + 2 * HW + p);
        float4 e3 = *(const float4*)(eb + 3 * HW + p);
        accum_var(l.x, e0.x, e1.x, e2.x, e3.x, sm, acc);
        accum_var(l.y, e0.y, e1.y, e2.y, e3.y, sm, acc);
        accum_var(l.z, e0.z, e1.z, e2.z, e3.z, sm, acc);
        accum_var(l.w, e0.w, e1.w, e2.w, e3.w, sm, acc);
    }

#pragma unroll
    for (int k = 0; k < NUM_LANES; ++k)
        for (int off = 16; off > 0; off >>= 1) acc[k] += __shfl_down(acc[k], off, 32);

    __shared__ float accs[NUM_LANES];
    if (threadIdx.x < NUM_LANES) accs[threadIdx.x] = 0.0f;
    __syncthreads();
    if ((threadIdx.x & 31) == 0) {
#pragma unroll
        for (int k = 0; k < NUM_LANES; ++k) atomicAdd(&accs[k], acc[k]);
    }
    __syncthreads();
    if (threadIdx.x < NUM_LANES)
        atomicAdd(&varsum[b * NUM_LANES + threadIdx.x], accs[threadIdx.x]);
}

// ---------------------------------------------------------------------------
// Kernel 4: faithful running-division scan -> two scalars
// ---------------------------------------------------------------------------
__global__ void dl_final_kernel(const float* __restrict__ counts,
                                const float* __restrict__ varsum,
                                const float* __restrict__ pvals, float* __restrict__ out) {
    if (threadIdx.x == 0 && blockIdx.x == 0) {
        float v = 0.0f, d = 0.0f;
        for (int b = 0; b < 8; ++b) {
            float s = 0.0f;
            for (int k = 0; k < NUM_LANES; ++k)
                s += varsum[b * NUM_LANES + k] / counts[b * NUM_LANES + k];
            v = (v + s) / (float)NUM_LANES;
            d = (d + pvals[b]) / (float)(2 * NUM_LANES * (NUM_LANES - 1));
        }
        out[0] = v / 8.0f;
        out[1] = d / 8.0f;
    }
}

// ---------------------------------------------------------------------------
extern "C" void kernel_launch(void* const* d_in, const int* in_sizes, int n_in,
                              void* d_out, int out_size, void* d_ws, size_t ws_size,
                              hipStream_t stream) {
    const float* emb = (const float*)d_in[0];
    const int* lab = (const int*)d_in[1];
    float* out = (float*)d_out;

    float* W = (float*)d_ws;
    float* counts = W + 0;    // 40
    float* sums   = W + 64;   // 160
    float* means  = W + 256;  // 160
    float* varsum = W + 448;  // 40
    float* pvals  = W + 512;  // 8

    dl_zero_ws<<<1, 1024, 0, stream>>>(W, 1024);

    dim3 grid(128, 8);
    dl_sums_kernel<<<grid, 256, 0, stream>>>(emb, lab, counts, sums);
    dl_means_gram_kernel<<<1, 32, 0, stream>>>(sums, counts, means, pvals);
    dl_var_kernel<<<grid, 256, 0, stream>>>(emb, lab, means, varsum);
    dl_final_kernel<<<1, 1, 0, stream>>>(counts, varsum, pvals, out);
}